// PEPEmbedding_47467978556122
// MI455X (gfx1250) — compile-verified
//
#include <hip/hip_runtime.h>
#include <hip/hip_bf16.h>
#include <math.h>

// PEP embedding: out[b,f,:] = sign(v[idx]) * relu(|v[idx]| - sigmoid(s[f,:]))
// with idx = x[b,f] + f*FIELD_SIZE.  Pure gather + elementwise: bandwidth-bound
// (~55 MB total @ 23.3 TB/s => ~2.4us floor).  No matrix op exists in this
// workload, so CDNA5-specific paths used are async global->LDS staging
// (ASYNCcnt) for the threshold tile, wave32 b128 row gathers, LDS b128
// threshold reads, and non-temporal b128 output stores.

#define FIELD_NUM      26
#define FIELD_SIZE     40000
#define LATENT_DIM     64
#define THR_ELEMS      (FIELD_NUM * LATENT_DIM)   // 1664 floats = 6656 B
#define THR_CHUNKS     (THR_ELEMS / 4)            // 416 float4 chunks
#define ROWS_PER_BLOCK 16                         // 16 lanes (b128 x4) per row
#define BLOCK_THREADS  256                        // 8 waves (wave32)

// native 128-bit vector (HIP's float4 is a struct -> rejected by
// __builtin_nontemporal_store)
typedef __attribute__((ext_vector_type(4))) float v4f;

// ---- kernel 1: thr = sigmoid(s) * GK (GK == 1.0), 1664 elements -----------
__global__ void pep_thr_kernel(const float* __restrict__ s,
                               float* __restrict__ thr, int n) {
  int i = blockIdx.x * blockDim.x + threadIdx.x;
  if (i < n) {
    thr[i] = 1.0f / (1.0f + expf(-s[i]));
  }
}

__device__ __forceinline__ float soft_thresh(float v, float t) {
  // sign(v) * relu(|v| - t); t > 0 so the v==0 case matches the reference.
  return copysignf(fmaxf(fabsf(v) - t, 0.0f), v);
}

// ---- kernel 2: gather + per-field soft threshold ---------------------------
__global__ void __launch_bounds__(BLOCK_THREADS)
pep_gather_kernel(const float* __restrict__ v,
                  const float* __restrict__ thr,   // [26*64] in workspace
                  const int*   __restrict__ x,     // [B*26] within-field ids
                  float*       __restrict__ out,   // [B*26*64]
                  int total_rows) {
  __shared__ float s_thr[THR_ELEMS];
  const int t = threadIdx.x;

  // Stage the 6656 B threshold tile global->LDS with gfx1250 async loads
  // (GVS mode: 64-bit SGPR base + 32-bit per-lane VGPR offset; VDST VGPR
  // carries the LDS byte offset == low 32 bits of the flat shared pointer).
  {
    const unsigned lds_base = (unsigned)(size_t)(const void*)s_thr;
    const unsigned goff0 = (unsigned)t * 16u;            // chunk t (t < 416)
    asm volatile("global_load_async_to_lds_b128 %0, %1, %2"
                 :: "v"(lds_base + goff0), "v"(goff0), "s"(thr)
                 : "memory");
    const unsigned c1 = (unsigned)t + BLOCK_THREADS;     // chunk t+256
    if (c1 < THR_CHUNKS) {
      const unsigned goff1 = c1 * 16u;
      asm volatile("global_load_async_to_lds_b128 %0, %1, %2"
                   :: "v"(lds_base + goff1), "v"(goff1), "s"(thr)
                   : "memory");
    }
    asm volatile("s_wait_asynccnt 0" ::: "memory");
  }
  __syncthreads();

  // 16 lanes cover one 64-float row with b128 each; 2 rows per wave32.
  const int row    = blockIdx.x * ROWS_PER_BLOCK + (t >> 4);
  const int lane16 = t & 15;
  if (row < total_rows) {
    const int f   = row % FIELD_NUM;                 // row = b*26 + f
    const int idx = x[row] + f * FIELD_SIZE;         // flat table row

    v4f val = ((const v4f*)(v + (size_t)idx * LATENT_DIM))[lane16];
    // conflict-free ds_load_b128: 16 lanes x 16 B = all 64 LDS banks
    v4f th  = ((const v4f*)(s_thr + f * LATENT_DIM))[lane16];

    val.x = soft_thresh(val.x, th.x);
    val.y = soft_thresh(val.y, th.y);
    val.z = soft_thresh(val.z, th.z);
    val.w = soft_thresh(val.w, th.w);

    // streaming write-once output: non-temporal b128 store
    __builtin_nontemporal_store(val,
        (v4f*)(out + (size_t)row * LATENT_DIM) + lane16);
  }
}

extern "C" void kernel_launch(void* const* d_in, const int* in_sizes, int n_in,
                              void* d_out, int out_size, void* d_ws, size_t ws_size,
                              hipStream_t stream) {
  const float* v = (const float*)d_in[0];   // [1,040,000 * 64] fp32
  const float* s = (const float*)d_in[1];   // [26 * 64] fp32
  const int*   x = (const int*)  d_in[2];   // [4096 * 26] int32
  float* out = (float*)d_out;               // [4096 * 26 * 64] fp32
  float* thr = (float*)d_ws;                // 1664 floats of scratch

  const int thr_n = in_sizes[1];            // 1664
  const int rows  = in_sizes[2];            // 4096*26 = 106,496

  pep_thr_kernel<<<(thr_n + 255) / 256, 256, 0, stream>>>(s, thr, thr_n);
  pep_gather_kernel<<<(rows + ROWS_PER_BLOCK - 1) / ROWS_PER_BLOCK,
                      BLOCK_THREADS, 0, stream>>>(v, thr, x, out, rows);
}